// RigidTransformation3DImputation_30270929502430
// MI455X (gfx1250) — compile-verified
//
#include <hip/hip_runtime.h>
#include <math.h>

// ---------------------------------------------------------------------------
// RigidTransformation3DImputation for MI455X (gfx1250).
// B=32, N=64, C=1.  Pipeline:
//   1) per-batch ZYZ rotation (angles identical for masks & image; image adds
//      translation c = -R*loc with loc rescaled by 64/66).
//   2) trilinear warp of m1/m2 (rotation only) -> d_out sections 1,2.
//   3) pad-by-1 "fill" warp of X -> X_t (zero border sampled implicitly).
//   4) out = real(ifft(F(X_t) . ifftshift(M1))) + real(ifft(F(Y) . ifftshift(M2)))
//      - fftshift/ifftshift eliminated: even-N shift == index XOR 32.
//      - each 1-D 64-pt (i)DFT done as a complex GEMM on V_WMMA_F32_16X16X4_F32
//        (f32 WMMA keeps FFT accuracy; problem is HBM-bound ~1.5GB/23.3TB/s).
//      - z-axis passes stage 64x16-complex panels into LDS via
//        GLOBAL_LOAD_ASYNC_TO_LDS_B128 (ASYNCcnt) when available.
// ---------------------------------------------------------------------------

#define PI_F 3.14159265358979323846f
#define NV3 262144            // 64^3
#define BATCH 32
#define VOLN (BATCH * NV3)

typedef float v2f __attribute__((ext_vector_type(2)));
typedef float v8f __attribute__((ext_vector_type(8)));
typedef int vi4 __attribute__((vector_size(16)));  // async-LDS builtin ptr type

#if __has_builtin(__builtin_amdgcn_wmma_f32_16x16x4_f32)
// D(16x16,f32) = A(16x4,f32) * B(4x16,f32) + C  -- CDNA5 wave32 WMMA
__device__ __forceinline__ v8f wmma_mac(v2f a, v2f b, v8f c) {
  return __builtin_amdgcn_wmma_f32_16x16x4_f32(false, a, false, b, (short)0, c,
                                               false, false);
}
#else
// Fallback so the host pass / other targets still compile. NOT the WMMA
// semantics -- only reached if the builtin is unavailable.
__device__ __forceinline__ v8f wmma_mac(v2f a, v2f b, v8f c) {
  c[0] += a.x * b.x + a.y * b.y;
  return c;
}
#endif

#if __has_builtin(__builtin_amdgcn_global_load_async_to_lds_b128) && \
    __has_builtin(__builtin_amdgcn_s_wait_asynccnt)
#define HAVE_ASYNC_LDS 1
#else
#define HAVE_ASYNC_LDS 0
#endif

__device__ __forceinline__ int iclamp(int v, int lo, int hi) {
  return v < lo ? lo : (v > hi ? hi : v);
}

// ---------------------------------------------------------------------------
// Per-batch affine params: P[b*12 + 0..8] = rm (row major), P[b*12 + 9..11] = c
// rm = (Rz(phi)*Ry(th)*Rz(psi))^T ; c = -rm * loc  (loc from rescaled theta_r)
// ---------------------------------------------------------------------------
__global__ void k_params(const float* __restrict__ theta, float* __restrict__ P) {
  int b = threadIdx.x;
  if (b >= BATCH) return;
  const float* th = theta + b * 6;
  float phi = th[0] * 2.f * PI_F - PI_F;
  float tht = th[1] * 2.f * PI_F - PI_F;
  float psi = th[2] * 2.f * PI_F - PI_F;
  float s1, c1, s2, c2, s3, c3;
  sincosf(phi, &s1, &c1);
  sincosf(tht, &s2, &c2);
  sincosf(psi, &s3, &c3);
  // M = Rz(phi)*Ry(th)*Rz(psi); rm = M^T
  float R0 = c1 * c2 * c3 - s1 * s3;   // M00
  float R1 = s1 * c2 * c3 + c1 * s3;   // M10
  float R2 = -s2 * c3;                 // M20
  float R3 = -c1 * c2 * s3 - s1 * c3;  // M01
  float R4 = -s1 * c2 * s3 + c1 * c3;  // M11
  float R5 = s2 * s3;                  // M21
  float R6 = c1 * s2;                  // M02
  float R7 = s1 * s2;                  // M12
  float R8 = c2;                       // M22
  const float sc = 64.0f / 66.0f;      // 'fill' padding translation rescale
  float lx = ((th[3] - 0.5f) * sc + 0.5f) * 2.f - 1.f;
  float ly = ((th[4] - 0.5f) * sc + 0.5f) * 2.f - 1.f;
  float lz = ((th[5] - 0.5f) * sc + 0.5f) * 2.f - 1.f;
  float* p = P + b * 12;
  p[0] = R0; p[1] = R1; p[2] = R2;
  p[3] = R3; p[4] = R4; p[5] = R5;
  p[6] = R6; p[7] = R7; p[8] = R8;
  p[9]  = -(R0 * lx + R1 * ly + R2 * lz);
  p[10] = -(R3 * lx + R4 * ly + R5 * lz);
  p[11] = -(R6 * lx + R7 * ly + R8 * lz);
}

// ---------------------------------------------------------------------------
// Warp masks m1,m2 (rotation only, clamp sampling on 64^3) -> d_out sec1/sec2
// ---------------------------------------------------------------------------
__global__ __launch_bounds__(256) void k_warp_masks(
    const float* __restrict__ m1, const float* __restrict__ m2,
    const float* __restrict__ P, float* __restrict__ o1, float* __restrict__ o2) {
  int idx = blockIdx.x * 256 + threadIdx.x;
  int b = idx >> 18;
  int r = idx & (NV3 - 1);
  int x = r >> 12, y = (r >> 6) & 63, z = r & 63;
  const float* p = P + b * 12;
  const float gs = 2.0f / 63.0f;
  float gx = -1.f + x * gs, gy = -1.f + y * gs, gz = -1.f + z * gs;
  float tx = p[0] * gx + p[1] * gy + p[2] * gz;
  float ty = p[3] * gx + p[4] * gy + p[5] * gz;
  float tz = p[6] * gx + p[7] * gy + p[8] * gz;
  float xs = (tx + 1.f) * 31.5f, ys = (ty + 1.f) * 31.5f, zs = (tz + 1.f) * 31.5f;
  int xf = (int)floorf(xs), yf = (int)floorf(ys), zf = (int)floorf(zs);
  int x0 = iclamp(xf, 0, 63), x1 = iclamp(xf + 1, 0, 63);
  int y0 = iclamp(yf, 0, 63), y1 = iclamp(yf + 1, 0, 63);
  int z0 = iclamp(zf, 0, 63), z1 = iclamp(zf + 1, 0, 63);
  float dx = xs - (float)x0, dy = ys - (float)y0, dz = zs - (float)z0;  // vs CLIPPED corner (ref quirk)
  float ox = 1.f - dx, oy = 1.f - dy, oz = 1.f - dz;
  int base = b << 18;
  int i000 = base + (x0 << 12) + (y0 << 6) + z0;
  int i001 = base + (x0 << 12) + (y0 << 6) + z1;
  int i010 = base + (x0 << 12) + (y1 << 6) + z0;
  int i011 = base + (x0 << 12) + (y1 << 6) + z1;
  int i100 = base + (x1 << 12) + (y0 << 6) + z0;
  int i101 = base + (x1 << 12) + (y0 << 6) + z1;
  int i110 = base + (x1 << 12) + (y1 << 6) + z0;
  int i111 = base + (x1 << 12) + (y1 << 6) + z1;
  float w000 = ox * oy * oz, w001 = ox * oy * dz, w010 = ox * dy * oz, w011 = ox * dy * dz;
  float w100 = dx * oy * oz, w101 = dx * oy * dz, w110 = dx * dy * oz, w111 = dx * dy * dz;
  o1[idx] = w000 * m1[i000] + w001 * m1[i001] + w010 * m1[i010] + w011 * m1[i011] +
            w100 * m1[i100] + w101 * m1[i101] + w110 * m1[i110] + w111 * m1[i111];
  o2[idx] = w000 * m2[i000] + w001 * m2[i001] + w010 * m2[i010] + w011 * m2[i011] +
            w100 * m2[i100] + w101 * m2[i101] + w110 * m2[i110] + w111 * m2[i111];
}

// ---------------------------------------------------------------------------
// Warp X with implicit zero pad (66^3 grid, output = cropped interior)
// ---------------------------------------------------------------------------
__device__ __forceinline__ float fetch_pad(const float* __restrict__ X, int base,
                                           int a, int b, int c) {
  bool in = (a >= 1) & (a <= 64) & (b >= 1) & (b <= 64) & (c >= 1) & (c <= 64);
  return in ? X[base + ((a - 1) << 12) + ((b - 1) << 6) + (c - 1)] : 0.f;
}

__global__ __launch_bounds__(256) void k_warp_img(const float* __restrict__ X,
                                                  const float* __restrict__ P,
                                                  float* __restrict__ XT) {
  int idx = blockIdx.x * 256 + threadIdx.x;
  int b = idx >> 18;
  int r = idx & (NV3 - 1);
  int x = r >> 12, y = (r >> 6) & 63, z = r & 63;
  const float* p = P + b * 12;
  const float gs = 2.0f / 65.0f;  // linspace(-1,1,66), interior index i+1
  float gx = -1.f + (x + 1) * gs, gy = -1.f + (y + 1) * gs, gz = -1.f + (z + 1) * gs;
  float tx = p[0] * gx + p[1] * gy + p[2] * gz + p[9];
  float ty = p[3] * gx + p[4] * gy + p[5] * gz + p[10];
  float tz = p[6] * gx + p[7] * gy + p[8] * gz + p[11];
  float xs = (tx + 1.f) * 32.5f, ys = (ty + 1.f) * 32.5f, zs = (tz + 1.f) * 32.5f;
  int xf = (int)floorf(xs), yf = (int)floorf(ys), zf = (int)floorf(zs);
  int x0 = iclamp(xf, 0, 65), x1 = iclamp(xf + 1, 0, 65);
  int y0 = iclamp(yf, 0, 65), y1 = iclamp(yf + 1, 0, 65);
  int z0 = iclamp(zf, 0, 65), z1 = iclamp(zf + 1, 0, 65);
  float dx = xs - (float)x0, dy = ys - (float)y0, dz = zs - (float)z0;
  float ox = 1.f - dx, oy = 1.f - dy, oz = 1.f - dz;
  int base = b << 18;
  float v = ox * oy * oz * fetch_pad(X, base, x0, y0, z0) +
            ox * oy * dz * fetch_pad(X, base, x0, y0, z1) +
            ox * dy * oz * fetch_pad(X, base, x0, y1, z0) +
            ox * dy * dz * fetch_pad(X, base, x0, y1, z1) +
            dx * oy * oz * fetch_pad(X, base, x1, y0, z0) +
            dx * oy * dz * fetch_pad(X, base, x1, y0, z1) +
            dx * dy * oz * fetch_pad(X, base, x1, y1, z0) +
            dx * dy * dz * fetch_pad(X, base, x1, y1, z1);
  XT[idx] = v;
}

// ---------------------------------------------------------------------------
// 1-D 64-point (i)DFT along one axis as complex GEMM via f32 WMMA.
//   out[k, line] = sum_n W[k,n] * in[n, line],  W[k,n] = tw[(k*n)&63]
//   tw[t] = scale * (cos(2pi t/64), dirsign*sin(2pi t/64))
// Block = 128 threads = 4 waves; wave w computes k-rows [16w,16w+16) of a
// 16-line tile. 256 blocks per volume. element addr (complex units):
//   addr = lv*NV3 + hi*S1 + lo*S0 + n*Sn
// IN_REAL: input is float (imag=0).  OUT_MODE: 0=complex, 1=real store, 2=real +=
// ZAXIS:   strides are (S0=64,S1=4096,Sn=1): lines scatter, n contiguous.
//          -> complex input staged through LDS (async global->LDS when
//             available), stores packed as float4.
// ---------------------------------------------------------------------------
#define PANEL_PITCH 66  // complex pitch per line: 16B aligned, bank-shifted

template <int IN_REAL, int OUT_MODE, int ZAXIS>
__global__ __launch_bounds__(128) void k_dft(const void* __restrict__ inp,
                                             void* __restrict__ outp,
                                             int S0, int S1, int Sn,
                                             float dirsign, float scale) {
  __shared__ float2 tw[64];
  __shared__ float2 panel[(ZAXIS && !IN_REAL) ? 16 * PANEL_PITCH : 1];
  int tid = threadIdx.x;
  if (tid < 64) {
    float ang = (2.f * PI_F / 64.f) * (float)tid;
    float sv, cv;
    sincosf(ang, &sv, &cv);
    tw[tid].x = cv * scale;
    tw[tid].y = dirsign * sv * scale;
  }

  int lv = blockIdx.x >> 8;      // local volume in this chunk
  int ntile = blockIdx.x & 255;  // 16-line tile, 256 per volume
  int wave = tid >> 5;
  int lane = tid & 31;
  int col = lane & 15;   // N column (and A row within M-tile)
  int khalf = lane >> 4; // selects K pair {0,1} vs {2,3} inside a K=4 step
  int m = (wave << 4) + col;           // global output row for A fragment
  int hi = ntile >> 2;
  int lo = ((ntile & 3) << 4) + col;
  int baseB = lv * NV3 + hi * S1 + lo * S0;

  const float2* in_c = (const float2*)inp;
  const float* in_r = (const float*)inp;

  if (ZAXIS && !IN_REAL) {
    // Stage 16 lines x 64 complex (each line contiguous, Sn==1) into LDS.
    // 128 threads: 8 threads/line, 4x 16B chunks each.
    int line = tid >> 3;
    int j = tid & 7;
    int gbase = lv * NV3 + hi * S1 + (((ntile & 3) << 4) + line) * S0;
    const float2* gp = in_c + gbase;
#if HAVE_ASYNC_LDS
#pragma unroll
    for (int c = 0; c < 4; c++) {
      int n = (j + c * 8) * 2;  // 2 complex per 16B chunk
      __builtin_amdgcn_global_load_async_to_lds_b128(
          (vi4*)(gp + n), (vi4*)&panel[line * PANEL_PITCH + n], 0, 0);
    }
    __builtin_amdgcn_s_wait_asynccnt(0);
#else
#pragma unroll
    for (int c = 0; c < 4; c++) {
      int n = (j + c * 8) * 2;
      float4 q = *(const float4*)(gp + n);
      *(float4*)&panel[line * PANEL_PITCH + n] = q;
    }
#endif
  }
  __syncthreads();

  v8f accRR = {}, accII = {}, accRI = {}, accIR = {};

#pragma unroll
  for (int s = 0; s < 16; s++) {
    int n0 = (s << 2) + (khalf << 1);
    float2 t0 = tw[(m * n0) & 63];
    float2 t1 = tw[(m * (n0 + 1)) & 63];
    v2f a_re = {t0.x, t1.x};
    v2f a_im = {t0.y, t1.y};
    if (IN_REAL) {
      v2f b_re;
      if (ZAXIS) {  // Sn==1 -> one 8B load for both n's
        float2 rr = *(const float2*)(in_r + baseB + n0);
        b_re = {rr.x, rr.y};
      } else {
        b_re = {in_r[baseB + n0 * Sn], in_r[baseB + (n0 + 1) * Sn]};
      }
      accRR = wmma_mac(a_re, b_re, accRR);
      if (OUT_MODE == 0) accIR = wmma_mac(a_im, b_re, accIR);
    } else {
      v2f b_re, b_im;
      if (ZAXIS) {  // read {re0,im0,re1,im1} in one 16B LDS load
        const float4* prow = (const float4*)&panel[col * PANEL_PITCH];
        float4 q = prow[n0 >> 1];
        b_re = {q.x, q.z};
        b_im = {q.y, q.w};
      } else {
        float2 c0 = in_c[baseB + n0 * Sn];
        float2 c1 = in_c[baseB + (n0 + 1) * Sn];
        b_re = {c0.x, c1.x};
        b_im = {c0.y, c1.y};
      }
      accRR = wmma_mac(a_re, b_re, accRR);
      accII = wmma_mac(a_im, b_im, accII);
      if (OUT_MODE == 0) {
        accRI = wmma_mac(a_re, b_im, accRI);
        accIR = wmma_mac(a_im, b_re, accIR);
      }
    }
  }

  // D layout: VGPR v holds (M = v + 8*khalf, N = col)
  if (OUT_MODE == 0) {
    float2* out_c = (float2*)outp;
    if (ZAXIS) {  // Sn==1: 8 consecutive complex per lane -> 4x float4 stores
      int kb = (wave << 4) + (khalf << 3);
      float4* op = (float4*)(out_c + baseB + kb);
#pragma unroll
      for (int v2 = 0; v2 < 4; v2++) {
        float4 o = {accRR[2 * v2] - accII[2 * v2], accRI[2 * v2] + accIR[2 * v2],
                    accRR[2 * v2 + 1] - accII[2 * v2 + 1],
                    accRI[2 * v2 + 1] + accIR[2 * v2 + 1]};
        op[v2] = o;
      }
    } else {
#pragma unroll
      for (int v = 0; v < 8; v++) {
        int k = (wave << 4) + v + (khalf << 3);
        float2 o;
        o.x = accRR[v] - accII[v];
        o.y = accRI[v] + accIR[v];
        out_c[baseB + k * Sn] = o;
      }
    }
  } else {
    float* out_r = (float*)outp;
#pragma unroll
    for (int v = 0; v < 8; v++) {
      int k = (wave << 4) + v + (khalf << 3);
      float val = accRR[v] - accII[v];
      if (OUT_MODE == 1)
        out_r[baseB + k * Sn] = val;
      else
        out_r[baseB + k * Sn] += val;
    }
  }
}

// ---------------------------------------------------------------------------
// F *= ifftshift(mask):  even-N ifftshift == index XOR 32 per axis
// ---------------------------------------------------------------------------
__global__ __launch_bounds__(256) void k_mulmask(float2* __restrict__ f,
                                                 const float* __restrict__ msk) {
  int idx = blockIdx.x * 256 + threadIdx.x;
  int b = idx >> 18;
  int r = idx & (NV3 - 1);
  int mr = r ^ ((32 << 12) | (32 << 6) | 32);
  float m = msk[(b << 18) | mr];
  float2 v = f[idx];
  v.x *= m;
  v.y *= m;
  f[idx] = v;
}

// ---------------------------------------------------------------------------
extern "C" void kernel_launch(void* const* d_in, const int* in_sizes, int n_in,
                              void* d_out, int out_size, void* d_ws, size_t ws_size,
                              hipStream_t stream) {
  (void)in_sizes; (void)n_in; (void)out_size;
  const float* X = (const float*)d_in[0];
  const float* Y = (const float*)d_in[1];
  const float* m1 = (const float*)d_in[2];
  const float* m2 = (const float*)d_in[3];
  const float* theta = (const float*)d_in[4];
  float* out0 = (float*)d_out;           // output volume
  float* out1 = out0 + (size_t)VOLN;     // M1_t
  float* out2 = out0 + (size_t)2 * VOLN; // M2_t

  char* ws = (char*)d_ws;
  float* P = (float*)ws;  // 32*12 floats
  // chunk volumes so 2 complex ping-pong buffers fit in ws
  const size_t perVol = (size_t)NV3 * 8;  // 2 MB complex per volume
  size_t avail = ws_size > 4096 ? ws_size - 4096 : 0;
  int G = (int)(avail / (2 * perVol));
  if (G < 1) G = 1;
  if (G > BATCH) G = BATCH;
  float2* W0 = (float2*)(ws + 4096);
  float2* W1 = (float2*)(ws + 4096 + (size_t)G * perVol);
  float* XT = (float*)W1;  // warped X (float) aliases W1 (consumed by pass 1)

  k_params<<<1, 32, 0, stream>>>(theta, P);
  k_warp_masks<<<VOLN / 256, 256, 0, stream>>>(m1, m2, P, out1, out2);

  // axis strides (complex units): z:(64,4096,1)  y:(1,4096,64)  x:(1,64,4096)
  const float FWD = -1.f, INV = 1.f, S = 1.f / 64.f;

  // ---- X branch: out0 = real(ifft(F(X_t) . M1s)) ----
  for (int v0 = 0; v0 < BATCH; v0 += G) {
    int g = (BATCH - v0) < G ? (BATCH - v0) : G;
    dim3 gD(g * 256), gP(g * (NV3 / 256));
    k_warp_img<<<gP, 256, 0, stream>>>(X + (size_t)v0 * NV3, P + v0 * 12, XT);
    k_dft<1, 0, 1><<<gD, 128, 0, stream>>>(XT, W0, 64, 4096, 1, FWD, 1.f);
    k_dft<0, 0, 0><<<gD, 128, 0, stream>>>(W0, W1, 1, 4096, 64, FWD, 1.f);
    k_dft<0, 0, 0><<<gD, 128, 0, stream>>>(W1, W0, 1, 64, 4096, FWD, 1.f);
    k_mulmask<<<gP, 256, 0, stream>>>(W0, out1 + (size_t)v0 * NV3);
    k_dft<0, 0, 1><<<gD, 128, 0, stream>>>(W0, W1, 64, 4096, 1, INV, S);
    k_dft<0, 0, 0><<<gD, 128, 0, stream>>>(W1, W0, 1, 4096, 64, INV, S);
    k_dft<0, 1, 0><<<gD, 128, 0, stream>>>(W0, out0 + (size_t)v0 * NV3, 1, 64, 4096, INV, S);
  }
  // ---- Y branch: out0 += real(ifft(F(Y) . M2s)) ----
  for (int v0 = 0; v0 < BATCH; v0 += G) {
    int g = (BATCH - v0) < G ? (BATCH - v0) : G;
    dim3 gD(g * 256), gP(g * (NV3 / 256));
    k_dft<1, 0, 1><<<gD, 128, 0, stream>>>(Y + (size_t)v0 * NV3, W0, 64, 4096, 1, FWD, 1.f);
    k_dft<0, 0, 0><<<gD, 128, 0, stream>>>(W0, W1, 1, 4096, 64, FWD, 1.f);
    k_dft<0, 0, 0><<<gD, 128, 0, stream>>>(W1, W0, 1, 64, 4096, FWD, 1.f);
    k_mulmask<<<gP, 256, 0, stream>>>(W0, out2 + (size_t)v0 * NV3);
    k_dft<0, 0, 1><<<gD, 128, 0, stream>>>(W0, W1, 64, 4096, 1, INV, S);
    k_dft<0, 0, 0><<<gD, 128, 0, stream>>>(W1, W0, 1, 4096, 64, INV, S);
    k_dft<0, 2, 0><<<gD, 128, 0, stream>>>(W0, out0 + (size_t)v0 * NV3, 1, 64, 4096, INV, S);
  }
}